// Model_47510928228872
// MI455X (gfx1250) — compile-verified
//
#include <hip/hip_runtime.h>

// CDNA5 / gfx1250: wave32, V_WMMA_F32_16X16X4_F32 for the H=3 LSTM recurrence.
// Transposed formulation D[gate,batch] = A_weights(16x4) x B_state(4x16) + C,
// 16 batch elements per wave. Gate rows are PERMUTED so each lane-pair owns
// complete channels (rows 0..7 = i0,i1,f0,f1,g0,g1,o0,o1 -> lower lane handles
// channels 0,1; rows 8..11 = i2,f2,g2,o2 -> upper lane handles channel 2),
// making the recurrence shuffle-free. The input-projection WMMA and the global
// loads are software-pipelined one step ahead so only the h-WMMA + activation
// chain sits on the serial dependency path. Per-lane masking of unused B
// elements uses multiply-by-{0,1} instead of selects so the compiler keeps
// the loads unconditional (no exec-mask branches in the loop).

typedef __attribute__((ext_vector_type(2))) float v2f;   // A/B operands: 2 VGPRs
typedef __attribute__((ext_vector_type(8))) float v8f;   // C/D: 8 VGPRs

#define TLEN 512

__device__ __forceinline__ float fsig(float x) {
  const float L2E = 1.4426950408889634f;
  return __builtin_amdgcn_rcpf(1.0f + __builtin_amdgcn_exp2f(-x * L2E));
}
__device__ __forceinline__ float ftanh(float x) {
  const float L2E = 1.4426950408889634f;
  return 2.0f * __builtin_amdgcn_rcpf(1.0f + __builtin_amdgcn_exp2f(-2.0f * L2E * x)) - 1.0f;
}

// Permuted gate row for A-row index n (0..11): rows 0..7 -> gates
// {0,1,3,4,6,7,9,10}, rows 8..11 -> gates {2,5,8,11}.
__device__ __forceinline__ int gate_of_row(int n) {
  return (n < 8) ? ((n >> 1) * 3 + (n & 1)) : ((n - 8) * 3 + 2);
}

// D layout: lane n holds rows 0..7 of column n; lane n+16 holds rows 8..15.
// Lower lane slots: A=channel0, B=channel1. Upper lane: A=channel2 (B unused).
__device__ __forceinline__ void cell_update(const v8f d, bool lo,
                                            float& cA, float& cB,
                                            float& hA, float& hB) {
  float iA = d[0];                  // lower: i0   upper: i2
  float iB = d[1];                  // lower: i1
  float fA = lo ? d[2] : d[1];      // lower: f0   upper: f2
  float fB = d[3];                  // lower: f1
  float gA = lo ? d[4] : d[2];      // lower: g0   upper: g2
  float gB = d[5];                  // lower: g1
  float oA = lo ? d[6] : d[3];      // lower: o0   upper: o2
  float oB = d[7];                  // lower: o1
  cA = fsig(fA) * cA + fsig(iA) * ftanh(gA);
  cB = fsig(fB) * cB + fsig(iB) * ftanh(gB);
  hA = fsig(oA) * ftanh(cA);
  hB = fsig(oB) * ftanh(cB);
}

// ---------------- Layer 0 (both directions), writes y0[B,T,6] -------------
__global__ void __launch_bounds__(256)
lstm_l0_kernel(const float* __restrict__ X,
               const float* __restrict__ WihF, const float* __restrict__ WhhF,
               const float* __restrict__ biF,  const float* __restrict__ bhF,
               const float* __restrict__ WihR, const float* __restrict__ WhhR,
               const float* __restrict__ biR,  const float* __restrict__ bhR,
               float* __restrict__ Y0) {
  const int lane = threadIdx.x & 31;
  const int n    = lane & 15;          // batch column / A row index
  const bool lo  = lane < 16;
  const int wv   = blockIdx.x * (blockDim.x >> 5) + (threadIdx.x >> 5);
  const int dir  = blockIdx.y;
  const int b    = wv * 16 + n;

  const float* Wi = dir ? WihR : WihF;
  const float* Wh = dir ? WhhR : WhhF;
  const float* bi = dir ? biR  : biF;
  const float* bh = dir ? bhR  : bhF;

  // A matrices (16x4 f32): lanes 0-15 row n hold K0,K1; lanes 16-31 hold K2,K3.
  // A_w K3 column carries the combined bias (paired with B_h K3 = 1.0).
  v2f ax = {0.f, 0.f}, aw = {0.f, 0.f};
  if (n < 12) {
    const int gm = gate_of_row(n);
    if (lo) { ax[0] = Wi[gm * 3 + 0]; ax[1] = Wi[gm * 3 + 1];
              aw[0] = Wh[gm * 3 + 0]; aw[1] = Wh[gm * 3 + 1]; }
    else    { ax[0] = Wi[gm * 3 + 2]; ax[1] = 0.f;
              aw[0] = Wh[gm * 3 + 2]; aw[1] = bi[gm] + bh[gm]; }
  }

  float hA = 0.f, hB = 0.f, cA = 0.f, cB = 0.f;
  const v8f zero = {};
  const int   xo0 = lo ? 0 : 2;        // branch-free per-lane load offsets
  const int   xo1 = lo ? 1 : 2;
  const float kM  = lo ? 1.f : 0.f;    // arithmetic lane mask (keeps loads
                                       // unconditional; data is finite)
  const int sx  = dir ? -3 : 3;        // x row stride (floats)
  const int sy  = dir ? -6 : 6;        // y0 row stride (floats)

  const float* pX = X + (size_t)b * TLEN * 3 + (dir ? (size_t)(TLEN - 1) * 3 : 0);
  float*       pY = Y0 + (size_t)b * TLEN * 6 + (dir ? (size_t)(TLEN - 1) * 6 : 0)
                       + 3 * dir + (lo ? 0 : 2);

  // Pipeline preamble: load t0 input and compute its projection.
  v2f bx;  bx[0] = pX[xo0];  bx[1] = pX[xo1] * kM;
  v8f acc = __builtin_amdgcn_wmma_f32_16x16x4_f32(false, ax, false, bx,
                                                  (short)0, zero, false, false);

  for (int step = 0; step < TLEN; ++step) {
    v2f bhv;  bhv[0] = hA;  bhv[1] = lo ? hB : 1.f;   // K0,K1=h0,h1 | K2,K3=h2,1
    v8f g = __builtin_amdgcn_wmma_f32_16x16x4_f32(false, aw, false, bhv,
                                                  (short)0, acc, false, false);
    // Prefetch next step's input + projection (overlaps activation chain).
    pX += (step + 1 < TLEN) ? sx : 0;                 // last iter: in-bounds dup
    bx[0] = pX[xo0];  bx[1] = pX[xo1] * kM;
    acc = __builtin_amdgcn_wmma_f32_16x16x4_f32(false, ax, false, bx,
                                                (short)0, zero, false, false);
    cell_update(g, lo, cA, cB, hA, hB);

    pY[0] = hA;                      // lower: h0 -> ch0, upper: h2 -> ch2
    if (lo) pY[1] = hB;              // h1 -> ch1
    pY += (step + 1 < TLEN) ? sy : 0;
  }
}

// ------- Layer 1 forward (T steps) + reverse (1 step) + linear + sigmoid ----
__global__ void __launch_bounds__(256)
lstm_l1_kernel(const float* __restrict__ Y0,
               const float* __restrict__ WiF, const float* __restrict__ WhF,
               const float* __restrict__ biF, const float* __restrict__ bhF,
               const float* __restrict__ WiR, const float* __restrict__ WhR,
               const float* __restrict__ biR, const float* __restrict__ bhR,
               const float* __restrict__ Wlin, const float* __restrict__ blin,
               float* __restrict__ out) {
  const int lane = threadIdx.x & 31;
  const int n    = lane & 15;
  const bool lo  = lane < 16;
  const int wv   = blockIdx.x * (blockDim.x >> 5) + (threadIdx.x >> 5);
  const int b    = wv * 16 + n;

  // Input dim = 6 -> two x-WMMAs (K=4 + K=2). Rows permuted via gate_of_row.
  v2f ax1 = {0.f, 0.f}, ax2 = {0.f, 0.f}, aw = {0.f, 0.f};
  v2f rx1 = {0.f, 0.f}, rx2 = {0.f, 0.f}, rw = {0.f, 0.f};
  if (n < 12) {
    const int gm = gate_of_row(n);
    if (lo) {
      ax1[0] = WiF[gm * 6 + 0]; ax1[1] = WiF[gm * 6 + 1];
      ax2[0] = WiF[gm * 6 + 4]; ax2[1] = WiF[gm * 6 + 5];
      aw[0]  = WhF[gm * 3 + 0]; aw[1]  = WhF[gm * 3 + 1];
      rx1[0] = WiR[gm * 6 + 0]; rx1[1] = WiR[gm * 6 + 1];
      rx2[0] = WiR[gm * 6 + 4]; rx2[1] = WiR[gm * 6 + 5];
      rw[0]  = WhR[gm * 3 + 0]; rw[1]  = WhR[gm * 3 + 1];
    } else {
      ax1[0] = WiF[gm * 6 + 2]; ax1[1] = WiF[gm * 6 + 3];
      aw[0]  = WhF[gm * 3 + 2]; aw[1]  = biF[gm] + bhF[gm];
      rx1[0] = WiR[gm * 6 + 2]; rx1[1] = WiR[gm * 6 + 3];
      rw[0]  = WhR[gm * 3 + 2]; rw[1]  = biR[gm] + bhR[gm];
    }
  }

  float hA = 0.f, hB = 0.f, cA = 0.f, cB = 0.f;
  const v8f zero = {};
  const int   yoA = lo ? 0 : 2;        // b64 load offsets (floats), 8B-aligned
  const int   yoB = lo ? 4 : 2;
  const float kM  = lo ? 1.f : 0.f;    // arithmetic lane mask

  const float* pY = Y0 + (size_t)b * TLEN * 6;

  // Pipeline preamble: t0 input + projection.
  float2 dA = *(const float2*)(pY + yoA);   // lower:y0,y1  upper:y2,y3
  float2 dB = *(const float2*)(pY + yoB);   // lower:y4,y5  upper:dup
  v2f bx1;  bx1[0] = dA.x;       bx1[1] = dA.y;
  v2f bx2;  bx2[0] = dB.x * kM;  bx2[1] = dB.y * kM;
  v8f acc = __builtin_amdgcn_wmma_f32_16x16x4_f32(false, ax1, false, bx1,
                                                  (short)0, zero, false, false);
  acc = __builtin_amdgcn_wmma_f32_16x16x4_f32(false, ax2, false, bx2,
                                              (short)0, acc, false, false);

  for (int t = 0; t < TLEN; ++t) {
    v2f bhv;  bhv[0] = hA;  bhv[1] = lo ? hB : 1.f;
    v8f g = __builtin_amdgcn_wmma_f32_16x16x4_f32(false, aw, false, bhv,
                                                  (short)0, acc, false, false);
    // Prefetch next step (last iter re-loads t=T-1: reused by reverse step).
    pY += (t + 1 < TLEN) ? 6 : 0;
    dA = *(const float2*)(pY + yoA);
    dB = *(const float2*)(pY + yoB);
    bx1[0] = dA.x;       bx1[1] = dA.y;
    bx2[0] = dB.x * kM;  bx2[1] = dB.y * kM;
    acc = __builtin_amdgcn_wmma_f32_16x16x4_f32(false, ax1, false, bx1,
                                                (short)0, zero, false, false);
    acc = __builtin_amdgcn_wmma_f32_16x16x4_f32(false, ax2, false, bx2,
                                                (short)0, acc, false, false);
    cell_update(g, lo, cA, cB, hA, hB);
  }
  // lower lane: hA,hB = h1f[0],h1f[1]; upper lane: hA = h1f[2].

  // Layer-1 reverse direction: y1[:, T-1, 3:6] is the reverse scan's FIRST
  // step (zero initial state) applied to y0[:, T-1, :]. bx1/bx2 still hold
  // the t=T-1 row from the final pipelined preload.
  float pSelf;
  {
    v8f racc = __builtin_amdgcn_wmma_f32_16x16x4_f32(false, rx1, false, bx1,
                                                     (short)0, zero, false, false);
    racc = __builtin_amdgcn_wmma_f32_16x16x4_f32(false, rx2, false, bx2,
                                                 (short)0, racc, false, false);
    v2f bhv;  bhv[0] = 0.f;  bhv[1] = lo ? 0.f : 1.f;  // h=0, bias column
    v8f g = __builtin_amdgcn_wmma_f32_16x16x4_f32(false, rw, false, bhv,
                                                  (short)0, racc, false, false);
    float rcA = 0.f, rcB = 0.f, rhA, rhB;
    cell_update(g, lo, rcA, rcB, rhA, rhB);

    // Partial dot with W_lin: lower covers channels 0,1 (fwd+rev); upper
    // covers channel 2 (fwd+rev). One shuffle combines the halves.
    pSelf = lo ? (Wlin[0] * hA + Wlin[1] * hB + Wlin[3] * rhA + Wlin[4] * rhB)
               : (Wlin[2] * hA + Wlin[5] * rhA);
  }
  float pUp = __shfl(pSelf, n | 16, 32);
  if (lo) out[b] = fsig(pSelf + pUp + blin[0]);
}

extern "C" void kernel_launch(void* const* d_in, const int* in_sizes, int n_in,
                              void* d_out, int out_size, void* d_ws, size_t ws_size,
                              hipStream_t stream) {
  const float* X        = (const float*)d_in[0];
  const float* W_ih_l0  = (const float*)d_in[1];
  const float* W_hh_l0  = (const float*)d_in[2];
  const float* b_ih_l0  = (const float*)d_in[3];
  const float* b_hh_l0  = (const float*)d_in[4];
  const float* W_ih_l0r = (const float*)d_in[5];
  const float* W_hh_l0r = (const float*)d_in[6];
  const float* b_ih_l0r = (const float*)d_in[7];
  const float* b_hh_l0r = (const float*)d_in[8];
  const float* W_ih_l1  = (const float*)d_in[9];
  const float* W_hh_l1  = (const float*)d_in[10];
  const float* b_ih_l1  = (const float*)d_in[11];
  const float* b_hh_l1  = (const float*)d_in[12];
  const float* W_ih_l1r = (const float*)d_in[13];
  const float* W_hh_l1r = (const float*)d_in[14];
  const float* b_ih_l1r = (const float*)d_in[15];
  const float* b_hh_l1r = (const float*)d_in[16];
  const float* W_lin    = (const float*)d_in[17];
  const float* b_lin    = (const float*)d_in[18];

  const int B = in_sizes[0] / (TLEN * 3);          // 8192
  float* Y0 = (float*)d_ws;                        // [B, T, 6] = 100.7 MB, L2-resident

  const int wavesPerBlock = 8;                     // 256 threads = 8 wave32
  const int blocks = B / (16 * wavesPerBlock);     // 64
  dim3 blk(256);

  lstm_l0_kernel<<<dim3(blocks, 2), blk, 0, stream>>>(
      X, W_ih_l0, W_hh_l0, b_ih_l0, b_hh_l0,
      W_ih_l0r, W_hh_l0r, b_ih_l0r, b_hh_l0r, Y0);

  lstm_l1_kernel<<<dim3(blocks), blk, 0, stream>>>(
      Y0, W_ih_l1, W_hh_l1, b_ih_l1, b_hh_l1,
      W_ih_l1r, W_hh_l1r, b_ih_l1r, b_hh_l1r,
      W_lin, b_lin, (float*)d_out);
}